// ConvolutionLayer_24876450579116
// MI455X (gfx1250) — compile-verified
//
#include <hip/hip_runtime.h>
#include <hip/hip_bf16.h>

typedef __attribute__((ext_vector_type(2))) float v2f;
typedef __attribute__((ext_vector_type(8))) float v8f;
typedef __attribute__((ext_vector_type(4))) unsigned int v4u;
typedef __attribute__((ext_vector_type(8))) int v8i;
typedef __attribute__((ext_vector_type(4))) int v4i;

constexpr int BB = 16;    // batches
constexpr int NN = 1024;  // nodes
constexpr int FF = 128;   // features in/out
constexpr int KC = 32;    // K chunk staged in LDS
constexpr int AST = 36;   // LDS stride A tiles: 32 + 4 pad (TDM pad_interval=4, pad_amount=3)
constexpr int BST = 132;  // LDS stride B tiles: 128 + 4 pad (TDM pad_interval=6, pad_amount=3)

// ---------------------------------------------------------------------------
// TDM 2-D tile load: global (row-major, stride0 elements/row) -> LDS with
// row padding. Descriptor per CDNA5 ISA 8.3/8.4. data_size = 4 bytes.
// ---------------------------------------------------------------------------
__device__ __forceinline__ void tdm_load_2d(unsigned ldsOff, unsigned long long gaddr,
                                            unsigned tile0, unsigned tile1,
                                            unsigned stride0, unsigned padInt,
                                            unsigned padAmt) {
  v4u g0;
  g0[0] = 1u;                                   // count=1 (valid user descriptor)
  g0[1] = ldsOff;                               // lds_addr (bytes)
  g0[2] = (unsigned)(gaddr & 0xffffffffu);      // global_addr[31:0]
  g0[3] = (unsigned)((gaddr >> 32) & 0x01ffffffu) | 0x80000000u;  // addr[56:32] | type=2

  const unsigned td0 = 1u << 20, td1 = 1u << 20;  // generous dims, tile fully in-bounds
  v8i g1;
  g1[0] = (int)((2u << 16) | (1u << 20) | (padInt << 22) | (padAmt << 25));  // dsz=4B, pad
  g1[1] = (int)((td0 & 0xffffu) << 16);                                  // tensor_dim0 lo16
  g1[2] = (int)(((td0 >> 16) & 0xffffu) | ((td1 & 0xffffu) << 16));      // dim0 hi | dim1 lo
  g1[3] = (int)(((td1 >> 16) & 0xffffu) | (tile0 << 16));                // dim1 hi | tile_dim0
  g1[4] = (int)(tile1 & 0xffffu);                                        // tile_dim1, tile_dim2=0
  g1[5] = (int)stride0;                                                  // dim0_stride lo32
  g1[6] = 0;                                                             // dim0_stride hi16
  g1[7] = 0;

  v4i z4 = (v4i)0;
#if defined(__clang_major__) && __clang_major__ >= 23
  v8i z8 = (v8i)0;
  __builtin_amdgcn_tensor_load_to_lds(g0, g1, z4, z4, z8, 0);
#else
  __builtin_amdgcn_tensor_load_to_lds(g0, g1, z4, z4, 0);
#endif
}

__device__ __forceinline__ unsigned lds_offset(const void* p) {
  return (unsigned)(unsigned long long)p;  // low 32 bits of generic addr == LDS byte offset
}

// ---------------------------------------------------------------------------
// Kernel 1:  Y[m, o] = sum_f X[m, f] * W[o, f]   (m = b*N + n, flattened)
// ---------------------------------------------------------------------------
__global__ __launch_bounds__(256) void gcn_linear_kernel(
    const float* __restrict__ X, const float* __restrict__ W,
    float* __restrict__ Y) {
  __shared__ float Xs[128 * AST];
  __shared__ float Ws[128 * AST];

  const int t = threadIdx.x;
  const int rowBase = blockIdx.x * 128;
  const int lane = t & 31, wv = t >> 5;
  const int l = lane & 15, h = lane >> 4;
  const int mb = wv * 16;

  v8f acc[8] = {};

  for (int k0 = 0; k0 < FF; k0 += KC) {
    __syncthreads();
#pragma unroll
    for (int i = 0; i < 4; ++i) {
      int idx = t + i * 256;
      int row = idx >> 3;
      int c4 = (idx & 7) << 2;
      *(float4*)(Xs + row * AST + c4) =
          *(const float4*)(X + (size_t)(rowBase + row) * FF + k0 + c4);
      *(float4*)(Ws + row * AST + c4) =
          *(const float4*)(W + (size_t)row * FF + k0 + c4);
    }
    __syncthreads();
#pragma unroll
    for (int k = 0; k < KC; k += 4) {
      v2f a;
      a.x = Xs[(mb + l) * AST + k + 2 * h];
      a.y = Xs[(mb + l) * AST + k + 2 * h + 1];
#pragma unroll
      for (int nt = 0; nt < 8; ++nt) {
        v2f bm;  // B[k][n] = W[n][k]
        bm.x = Ws[(nt * 16 + l) * AST + k + 2 * h];
        bm.y = Ws[(nt * 16 + l) * AST + k + 2 * h + 1];
        acc[nt] = __builtin_amdgcn_wmma_f32_16x16x4_f32(
            false, a, false, bm, (short)0, acc[nt], false, false);
      }
    }
  }

#pragma unroll
  for (int nt = 0; nt < 8; ++nt)
#pragma unroll
    for (int r = 0; r < 8; ++r) {
      int row = rowBase + mb + r + 8 * h;
      int col = nt * 16 + l;
      Y[(size_t)row * FF + col] = acc[nt][r];
    }
}

// ---------------------------------------------------------------------------
// Kernel 2:  out[b,n,o] = lrelu( (adj[b,n,:] @ Y[b,:,o]) / deg[b,n] + bias[o] )
// TDM-staged, double-buffered tiles; deg folded from LDS-resident adjacency.
// ---------------------------------------------------------------------------
__global__ __launch_bounds__(256) void gcn_agg_kernel(
    const float* __restrict__ adj, const float* __restrict__ Y,
    const float* __restrict__ bias, float* __restrict__ out) {
  __shared__ float As[2][128 * AST];
  __shared__ float Bs[2][KC * BST];
  __shared__ float degs[128];

  const int t = threadIdx.x;
  const int rowBase = blockIdx.x * 128;
  const int b = blockIdx.y;
  const int lane = t & 31, wv = t >> 5;
  const int l = lane & 15, h = lane >> 4;
  const int mb = wv * 16;

  const float* adjB = adj + (size_t)b * NN * NN;
  const float* Yb = Y + (size_t)b * NN * FF;

  v8f acc[8] = {};
  float dacc = 0.0f;

  // prologue: DMA chunk 0 into buffer 0 (TDM ignores EXEC; one lane issues)
  if (t == 0) {
    tdm_load_2d(lds_offset(&As[0][0]),
                (unsigned long long)(adjB + (size_t)rowBase * NN),
                /*tile0=*/KC, /*tile1=*/128, /*stride0=*/NN, /*padInt=*/4, /*padAmt=*/3);
    tdm_load_2d(lds_offset(&Bs[0][0]),
                (unsigned long long)Yb,
                /*tile0=*/FF, /*tile1=*/KC, /*stride0=*/FF, /*padInt=*/6, /*padAmt=*/3);
  }
  __builtin_amdgcn_s_wait_tensorcnt(0);
  __syncthreads();

  constexpr int NCHUNK = NN / KC;
  for (int c = 0; c < NCHUNK; ++c) {
    const int cur = c & 1;
    // kick off next chunk's DMA into the other buffer while we compute
    if (t == 0 && c + 1 < NCHUNK) {
      const int k0n = (c + 1) * KC;
      tdm_load_2d(lds_offset(&As[cur ^ 1][0]),
                  (unsigned long long)(adjB + (size_t)rowBase * NN + k0n),
                  KC, 128, NN, 4, 3);
      tdm_load_2d(lds_offset(&Bs[cur ^ 1][0]),
                  (unsigned long long)(Yb + (size_t)k0n * FF),
                  FF, KC, FF, 6, 3);
    }

    if (t < 128) {  // fold row-degree from the LDS-resident adjacency tile
      float s = 0.0f;
#pragma unroll
      for (int kk = 0; kk < KC; ++kk) s += As[cur][t * AST + kk];
      dacc += s;
    }

#pragma unroll
    for (int k = 0; k < KC; k += 4) {
      v2f a;
      a.x = As[cur][(mb + l) * AST + k + 2 * h];
      a.y = As[cur][(mb + l) * AST + k + 2 * h + 1];
#pragma unroll
      for (int nt = 0; nt < 8; ++nt) {
        v2f bm;  // B[k][n] = Y[k0+k][n]
        bm.x = Bs[cur][(k + 2 * h) * BST + nt * 16 + l];
        bm.y = Bs[cur][(k + 2 * h + 1) * BST + nt * 16 + l];
        acc[nt] = __builtin_amdgcn_wmma_f32_16x16x4_f32(
            false, a, false, bm, (short)0, acc[nt], false, false);
      }
    }

    __builtin_amdgcn_s_wait_tensorcnt(0);  // next buffer landed (others have cnt==0)
    __syncthreads();                       // everyone done reading cur before reuse
  }

  if (t < 128) degs[t] = dacc;
  __syncthreads();

#pragma unroll
  for (int nt = 0; nt < 8; ++nt) {
    int col = nt * 16 + l;
    float bv = bias[col];
#pragma unroll
    for (int r = 0; r < 8; ++r) {
      int rl = mb + r + 8 * h;
      float v = acc[nt][r] / degs[rl] + bv;
      v = (v >= 0.0f) ? v : 0.01f * v;
      out[((size_t)b * NN + rowBase + rl) * FF + col] = v;
    }
  }
}

extern "C" void kernel_launch(void* const* d_in, const int* in_sizes, int n_in,
                              void* d_out, int out_size, void* d_ws, size_t ws_size,
                              hipStream_t stream) {
  const float* node = (const float*)d_in[0];  // [16,1024,128]
  const float* adj  = (const float*)d_in[1];  // [16,1024,1024]
  const float* W    = (const float*)d_in[2];  // [128,128]
  const float* bias = (const float*)d_in[3];  // [128]
  float* out = (float*)d_out;                 // [16,1024,128]
  float* Yws = (float*)d_ws;                  // scratch: [16*1024,128] fp32 = 8 MB

  dim3 blk(256);
  gcn_linear_kernel<<<dim3((BB * NN) / 128), blk, 0, stream>>>(node, W, Yws);
  gcn_agg_kernel<<<dim3(NN / 128, BB), blk, 0, stream>>>(adj, Yws, bias, out);
}